// SinkhornCorr_73847667687778
// MI455X (gfx1250) — compile-verified
//
#include <hip/hip_runtime.h>
#include <hip/hip_bf16.h>
#include <math.h>

#define NB 8
#define NPT 1536
#define EPS 1e-3f
#define INV_EPS 1000.0f
#define ITERS 100

typedef __attribute__((ext_vector_type(2))) float v2f;
typedef __attribute__((ext_vector_type(8))) float v8f;

// d_out layout (floats), matching reference tuple (x1, x2, P, idx12, idx21):
#define OFF_X1 0
#define OFF_X2 (NB*NPT*3)
#define OFF_P  (2*NB*NPT*3)
#define OFF_I12 (OFF_P + NB*NPT*NPT)
#define OFF_I21 (OFF_I12 + NB*NPT)

__global__ void sk_init_uv(float* uv) {
    int i = blockIdx.x * blockDim.x + threadIdx.x;
    if (i < 2 * NB * NPT) uv[i] = 0.0f;
}

__global__ void sk_copy_x(const float* __restrict__ x1, const float* __restrict__ x2,
                          float* __restrict__ out) {
    int i = blockIdx.x * blockDim.x + threadIdx.x;
    if (i < NB * NPT * 3) {
        out[OFF_X1 + i] = x1[i];
        out[OFF_X2 + i] = x2[i];
    }
}

// Build M[k,i,j] = ||x1[k,i] - x2[k,j]|| with V_WMMA_F32_16X16X4_F32.
// One wave computes one 16x16 tile: dot products via WMMA (K=3 padded to 4),
// norms computed in-lane and broadcast with __shfl.
__global__ void __launch_bounds__(128)
sk_build_m(const float* __restrict__ x1, const float* __restrict__ x2,
           float* __restrict__ M) {
    const int lane = threadIdx.x;        // 0..31, full wave active (EXEC all 1s)
    const int half = lane >> 4;          // 0: K=0/1, 1: K=2/3
    const int l16  = lane & 15;
    const int tj = (blockIdx.x * 4 + threadIdx.y) * 16;
    const int ti = blockIdx.y * 16;
    const int k  = blockIdx.z;

    const float* p = x1 + (size_t)(k * NPT + ti + l16) * 3;
    const float* q = x2 + (size_t)(k * NPT + tj + l16) * 3;
    float px = p[0], py = p[1], pz = p[2];
    float qx = q[0], qy = q[1], qz = q[2];
    float n1l = px * px + py * py + pz * pz;   // norm of x1 row (ti + lane&15)
    float n2l = qx * qx + qy * qy + qz * qz;   // norm of x2 col (tj + lane&15)

    // A: 16x4 (M=rows of x1, K=xyz0). VGPR0 = K0|K2, VGPR1 = K1|K3 per half-wave.
    v2f a, b;
    a.x = half ? pz : px;
    a.y = half ? 0.0f : py;
    // B: 4x16 (K=xyz0, N=cols of x2). Same K split across half-waves.
    b.x = half ? qz : qx;
    b.y = half ? 0.0f : qy;

    v8f c = {};
    c = __builtin_amdgcn_wmma_f32_16x16x4_f32(false, a, false, b, (short)0, c,
                                              false, false);

    // D layout: VGPR r -> row (r + 8*half), lane%16 -> col
    float* Mbase = M + (size_t)k * NPT * NPT + (size_t)(tj + l16);
    #pragma unroll
    for (int r = 0; r < 8; ++r) {
        float n1r = __shfl(n1l, r + 8 * half, 32);
        float d2 = n1r + n2l - 2.0f * c[r];
        float dist = sqrtf(fmaxf(d2, 0.0f));
        int i = ti + r + 8 * half;
        Mbase[(size_t)i * NPT] = dist;
    }
}

// u[k,i] = eps*(log a) - eps*LSE_j((v[k,j]-M[k,i,j])/eps). One wave per row,
// coalesced along the contiguous row; two-pass (max then sum) stabilized LSE.
__global__ void __launch_bounds__(256)
sk_row_pass(const float* __restrict__ M, const float* __restrict__ v,
            float* __restrict__ u) {
    const int lane = threadIdx.x;
    const int row = blockIdx.x * 8 + threadIdx.y;    // k*NPT + i
    const float* Mr = M + (size_t)row * NPT;
    const float* vk = v + (row / NPT) * NPT;

    float mx = -3.0e38f;
    for (int j = lane; j < NPT; j += 32) mx = fmaxf(mx, vk[j] - Mr[j]);
    #pragma unroll
    for (int off = 16; off; off >>= 1) mx = fmaxf(mx, __shfl_xor(mx, off, 32));

    float s = 0.0f;
    for (int j = lane; j < NPT; j += 32)
        s += __expf((vk[j] - Mr[j] - mx) * INV_EPS);
    #pragma unroll
    for (int off = 16; off; off >>= 1) s += __shfl_xor(s, off, 32);

    if (lane == 0)
        u[row] = EPS * (-logf((float)NB)) - mx - EPS * logf(s);
}

// v[k,j] = eps*(log b) - eps*LSE_i((u[k,i]-M[k,i,j])/eps). One thread per
// column so adjacent threads read adjacent j -> coalesced; M is L2-resident.
__global__ void __launch_bounds__(256)
sk_col_pass(const float* __restrict__ M, const float* __restrict__ u,
            float* __restrict__ v) {
    const int idx = blockIdx.x * 256 + threadIdx.x;  // k*NPT + j
    const int k = idx / NPT, j = idx % NPT;
    const float* Mc = M + (size_t)k * NPT * NPT + j;
    const float* uk = u + k * NPT;

    float mx = -3.0e38f;
    for (int i = 0; i < NPT; ++i) mx = fmaxf(mx, uk[i] - Mc[(size_t)i * NPT]);
    float s = 0.0f;
    for (int i = 0; i < NPT; ++i)
        s += __expf((uk[i] - Mc[(size_t)i * NPT] - mx) * INV_EPS);

    v[idx] = EPS * (-logf((float)NB)) - mx - EPS * logf(s);
}

// argmax_j P = argmax_j (v[j] - M[i,j]) (exp is monotonic, u_i constant/row).
// First-index tie-break to match jnp.argmax.
__global__ void __launch_bounds__(256)
sk_argmax_row(const float* __restrict__ M, const float* __restrict__ v,
              float* __restrict__ out12) {
    const int lane = threadIdx.x;
    const int row = blockIdx.x * 8 + threadIdx.y;
    const float* Mr = M + (size_t)row * NPT;
    const float* vk = v + (row / NPT) * NPT;

    float best = -3.0e38f; int bj = 0;
    for (int j = lane; j < NPT; j += 32) {
        float sc = vk[j] - Mr[j];
        if (sc > best) { best = sc; bj = j; }    // strict > keeps first index
    }
    #pragma unroll
    for (int off = 16; off; off >>= 1) {
        float ob = __shfl_xor(best, off, 32);
        int   oj = __shfl_xor(bj, off, 32);
        if (ob > best || (ob == best && oj < bj)) { best = ob; bj = oj; }
    }
    if (lane == 0) out12[row] = (float)bj;
}

__global__ void __launch_bounds__(256)
sk_argmax_col(const float* __restrict__ M, const float* __restrict__ u,
              float* __restrict__ out21) {
    const int idx = blockIdx.x * 256 + threadIdx.x;
    const int k = idx / NPT, j = idx % NPT;
    const float* Mc = M + (size_t)k * NPT * NPT + j;
    const float* uk = u + k * NPT;

    float best = -3.0e38f; int bi = 0;
    for (int i = 0; i < NPT; ++i) {
        float sc = uk[i] - Mc[(size_t)i * NPT];
        if (sc > best) { best = sc; bi = i; }
    }
    out21[idx] = (float)bi;
}

// In-place: overwrite M region with P = exp((u+v-M)/eps).
__global__ void __launch_bounds__(256)
sk_make_p(float* __restrict__ P, const float* __restrict__ u,
          const float* __restrict__ v) {
    size_t idx = (size_t)blockIdx.x * 256 + threadIdx.x;
    if (idx >= (size_t)NB * NPT * NPT) return;
    size_t kn = idx / NPT;           // k*NPT + i
    int j = (int)(idx % NPT);
    int k = (int)(kn / NPT);
    P[idx] = __expf((u[kn] + v[(size_t)k * NPT + j] - P[idx]) * INV_EPS);
}

extern "C" void kernel_launch(void* const* d_in, const int* in_sizes, int n_in,
                              void* d_out, int out_size, void* d_ws, size_t ws_size,
                              hipStream_t stream) {
    const float* x1 = (const float*)d_in[0];
    const float* x2 = (const float*)d_in[1];
    float* out = (float*)d_out;
    float* u = (float*)d_ws;          // NB*NPT floats
    float* v = u + NB * NPT;          // NB*NPT floats
    float* M = out + OFF_P;           // 75.5 MB scratch == final P region (L2-resident)

    sk_init_uv<<<(2 * NB * NPT + 255) / 256, 256, 0, stream>>>(u);
    sk_copy_x<<<(NB * NPT * 3 + 255) / 256, 256, 0, stream>>>(x1, x2, out);

    dim3 bB(32, 4), gB(NPT / 16 / 4, NPT / 16, NB);
    sk_build_m<<<gB, bB, 0, stream>>>(x1, x2, M);

    dim3 bR(32, 8);
    const int gR = NB * NPT / 8;       // wave per row
    const int gC = NB * NPT / 256;     // thread per column
    for (int it = 0; it < ITERS; ++it) {
        sk_row_pass<<<gR, bR, 0, stream>>>(M, v, u);
        sk_col_pass<<<gC, 256, 0, stream>>>(M, u, v);
    }

    // Argmaxes read M (scores), so run them before M is overwritten by P.
    sk_argmax_row<<<gR, bR, 0, stream>>>(M, v, out + OFF_I12);
    sk_argmax_col<<<gC, 256, 0, stream>>>(M, u, out + OFF_I21);

    size_t total = (size_t)NB * NPT * NPT;
    sk_make_p<<<(unsigned)((total + 255) / 256), 256, 0, stream>>>(M, u, v);
}